// MoELayer_22101901705637
// MI455X (gfx1250) — compile-verified
//
#include <hip/hip_runtime.h>
#include <math.h>

typedef __attribute__((ext_vector_type(2))) float v2f;
typedef __attribute__((ext_vector_type(4))) float v4f;
typedef __attribute__((ext_vector_type(8))) float v8f;

#define NB   16
#define LL   1024
#define DD   512
#define EE   8
#define DFFC 2048

#define MT      64    // L-rows per workgroup
#define KC      64    // DFF chunk width
#define NCHUNK  (DFFC / KC)
#define XSTRIDE 516   // padded LDS stride for x tile (bank-conflict free)
#define HSTRIDE 68    // padded LDS stride for h chunk

// ---------------- Router: logits, top-1, balance loss ----------------
__global__ __launch_bounds__(32) void router_kernel(
    const int* __restrict__ view_ids, const int* __restrict__ visit_ids,
    const float* __restrict__ rview, const float* __restrict__ rvisit,
    int* __restrict__ top1, float* __restrict__ loss_out) {
  __shared__ float probs[NB][EE];
  const int t = threadIdx.x;
  if (t < NB) {
    const int vi = view_ids[t];
    const int bi = visit_ids[t];
    float lg[EE];
    float mx = -1e30f;
    int arg = 0;
    for (int e = 0; e < EE; ++e) {
      lg[e] = rview[vi * EE + e] + rvisit[bi * EE + e];
      if (lg[e] > mx) { mx = lg[e]; arg = e; }
    }
    top1[t] = arg;
    float s = 0.f;
    for (int e = 0; e < EE; ++e) { float p = __expf(lg[e] - mx); probs[t][e] = p; s += p; }
    float inv = 1.f / s;
    for (int e = 0; e < EE; ++e) probs[t][e] *= inv;
  }
  __syncthreads();
  if (t == 0) {
    float loss = 0.f;
    for (int e = 0; e < EE; ++e) {
      float load = 0.f;
      for (int b = 0; b < NB; ++b) load += probs[b][e];
      load *= (1.f / NB);
      loss -= load * __logf(load);
    }
    loss_out[0] = loss;
  }
}

// ---------------- Fused expert FFN: out = relu(x*W1+b1)*W2 + b2 ----------------
// Wave mapping (8 waves): rtp = wave&1 -> rows [rtp*32, rtp*32+32)
//                         ctq = wave>>1 -> out cols [ctq*128, ctq*128+128)
// Each W fragment is reused for 2 row-tiles in-register (halves L2 W-traffic).
__global__ __launch_bounds__(256) void moe_ffn_kernel(
    const float* __restrict__ x, const int* __restrict__ top1,
    const float* __restrict__ W1, const float* __restrict__ b1,
    const float* __restrict__ W2, const float* __restrict__ b2,
    float* __restrict__ out) {
  __shared__ float lds_x[MT * XSTRIDE];   // 64 x 512 (padded)  ~132 KB
  __shared__ float lds_h[MT * HSTRIDE];   // 64 x 64  (padded)  ~17 KB

  const int b    = blockIdx.x;
  const int row0 = blockIdx.y * MT;
  const int tid  = threadIdx.x;
  const int lane = tid & 31;
  const int wave = tid >> 5;
  const int half = lane >> 4;   // 0: lanes 0-15, 1: lanes 16-31
  const int lm   = lane & 15;
  const int rtp  = wave & 1;    // row-tile pair (32 rows)
  const int ctq  = wave >> 1;   // output-column quarter (128 cols)

  const int e = top1[b];
  const float* xb  = x  + ((size_t)b * LL + row0) * DD;
  const float* w1e = W1 + (size_t)e * DD * DFFC;
  const float* w2e = W2 + (size_t)e * DFFC * DD;
  const float* b1e = b1 + (size_t)e * DFFC;
  const float* b2e = b2 + (size_t)e * DD;

  // Stage x tile into LDS (float4 loads, padded stride stores)
  for (int idx = tid * 4; idx < MT * DD; idx += 256 * 4) {
    const int r  = idx >> 9;         // / 512
    const int cc = idx & (DD - 1);
    v4f v = *(const v4f*)(xb + (size_t)r * DD + cc);
    *(v4f*)(&lds_x[r * XSTRIDE + cc]) = v;
  }
  __syncthreads();

  const v8f vzero = {};
  v8f acc0[8], acc1[8];
#pragma unroll
  for (int t = 0; t < 8; ++t) { acc0[t] = vzero; acc1[t] = vzero; }

  // A-fragment bases (16x4 f32 layout: lanes0-15 K=k0..k0+1, lanes16-31 K=k0+2..k0+3)
  const float* a_base0  = &lds_x[(rtp * 32 + lm) * XSTRIDE + 2 * half];
  const float* a_base1  = a_base0 + 16 * XSTRIDE;
  const float* ha_base0 = &lds_h[(rtp * 32 + lm) * HSTRIDE + 2 * half];
  const float* ha_base1 = ha_base0 + 16 * HSTRIDE;

  for (int c = 0; c < NCHUNK; ++c) {
    // Warm L2/WGP$ for the next chunk's weights while we compute this one.
    if (c + 1 < NCHUNK) {
      __builtin_prefetch(w1e + (size_t)(lane * 16) * DFFC + (c + 1) * KC + ctq * 16, 0, 1);
      __builtin_prefetch(w2e + (size_t)((c + 1) * KC + 2 * lane) * DD + ctq * 128, 0, 1);
    }

    // ---- stage 1: h tiles (rt0,hc=ctq),(rt1,hc=ctq) = relu(x * W1[:,cols] + b1) ----
    v8f h0 = vzero, h1 = vzero;
    const int n0 = c * KC + ctq * 16;  // this wave's 16-col h slice
    {
      const float* bp = w1e + (size_t)(2 * half) * DFFC + n0 + lm;
#pragma unroll 4
      for (int k0 = 0; k0 < DD; k0 += 4) {
        v2f a0 = *(const v2f*)(a_base0 + k0);
        v2f a1 = *(const v2f*)(a_base1 + k0);
        const float* bpk = bp + (size_t)k0 * DFFC;
        v2f bb;
        bb.x = bpk[0];
        bb.y = bpk[DFFC];
        h0 = __builtin_amdgcn_wmma_f32_16x16x4_f32(false, a0, false, bb, (short)0, h0, false, false);
        h1 = __builtin_amdgcn_wmma_f32_16x16x4_f32(false, a1, false, bb, (short)0, h1, false, false);
      }
    }
    {
      const float bv = b1e[n0 + lm];
      float* hw0 = &lds_h[(rtp * 32 + 8 * half) * HSTRIDE + ctq * 16 + lm];
      float* hw1 = hw0 + 16 * HSTRIDE;
#pragma unroll
      for (int v = 0; v < 8; ++v) {
        hw0[v * HSTRIDE] = fmaxf(h0[v] + bv, 0.f);
        hw1[v * HSTRIDE] = fmaxf(h1[v] + bv, 0.f);
      }
    }
    __syncthreads();

    // ---- stage 2: acc += h_chunk * W2[c*64 : c*64+64, cols] ----
    {
      const float* bp2 = w2e + (size_t)(c * KC + 2 * half) * DD + ctq * 128 + lm;
#pragma unroll 2
      for (int k0 = 0; k0 < KC; k0 += 4) {
        v2f a0 = *(const v2f*)(ha_base0 + k0);
        v2f a1 = *(const v2f*)(ha_base1 + k0);
        const float* bpk = bp2 + (size_t)k0 * DD;
#pragma unroll
        for (int t = 0; t < 8; ++t) {
          v2f bb;
          bb.x = bpk[t * 16];
          bb.y = bpk[DD + t * 16];
          acc0[t] = __builtin_amdgcn_wmma_f32_16x16x4_f32(false, a0, false, bb, (short)0, acc0[t], false, false);
          acc1[t] = __builtin_amdgcn_wmma_f32_16x16x4_f32(false, a1, false, bb, (short)0, acc1[t], false, false);
        }
      }
    }
    __syncthreads();
  }

  // ---- epilogue: + b2, store ----
  float* ob = out + ((size_t)b * LL + row0) * DD;
#pragma unroll
  for (int t = 0; t < 8; ++t) {
    const int col = ctq * 128 + t * 16 + lm;
    const float bias = b2e[col];
#pragma unroll
    for (int v = 0; v < 8; ++v) {
      const int r = rtp * 32 + v + 8 * half;
      ob[(size_t)r * DD + col]        = acc0[t][v] + bias;
      ob[(size_t)(r + 16) * DD + col] = acc1[t][v] + bias;
    }
  }
}

extern "C" void kernel_launch(void* const* d_in, const int* in_sizes, int n_in,
                              void* d_out, int out_size, void* d_ws, size_t ws_size,
                              hipStream_t stream) {
  const float* x         = (const float*)d_in[0];
  const int*   view_ids  = (const int*)d_in[1];
  const int*   visit_ids = (const int*)d_in[2];
  const float* rview     = (const float*)d_in[3];
  const float* rvisit    = (const float*)d_in[4];
  const float* W1        = (const float*)d_in[5];
  const float* b1        = (const float*)d_in[6];
  const float* W2        = (const float*)d_in[7];
  const float* b2        = (const float*)d_in[8];

  float* out  = (float*)d_out;
  int*   top1 = (int*)d_ws;
  float* loss = out + (size_t)NB * LL * DD;  // scalar appended after (B,L,D) output

  router_kernel<<<1, 32, 0, stream>>>(view_ids, visit_ids, rview, rvisit, top1, loss);

  dim3 grid(NB, LL / MT);
  moe_ffn_kernel<<<grid, 256, 0, stream>>>(x, top1, W1, b1, W2, b2, out);
}